// FullNet_55276229099597
// MI455X (gfx1250) — compile-verified
//
#include <hip/hip_runtime.h>

// Two-stream radiative transfer scan (up + down) for MI455X / gfx1250.
// B x C independent columns, serial over L. Memory-bound (~378 MB @ 23.3 TB/s).
// Strategy: per-layer 2-float carry checkpoint in LDS (big 320KB/WGP LDS),
// async global->LDS double-buffered input staging (ASYNCcnt), wave32
// width-16 shuffle reduction for the channel sums.

#define EPSF 1e-6f
constexpr int Bn = 2048, Ln = 60, Cn = 48;
constexpr int NB = 2;            // batches per block
constexpr int NT = Cn * NB;      // 96 threads = 3 full wave32s

#if defined(__has_builtin)
# if __has_builtin(__builtin_amdgcn_global_load_async_to_lds_b32)
#  define ASYNC_LDS 1
# else
#  define ASYNC_LDS 0
# endif
#else
# define ASYNC_LDS 0
#endif

#if ASYNC_LDS
# if __has_builtin(__builtin_amdgcn_s_wait_asynccnt)
#  define WAIT_ASYNC0() __builtin_amdgcn_s_wait_asynccnt(0)
# else
#  define WAIT_ASYNC0() asm volatile("s_wait_asynccnt 0x0" ::: "memory")
# endif
typedef __attribute__((address_space(1))) int Gint;   // global (prints as __device__ int*)
typedef __attribute__((address_space(3))) int Lint;   // LDS    (prints as __shared__ int*)
#endif

__device__ __forceinline__ float red16(float v) {
  // Butterfly sum across a 16-lane group (wave32, width 16).
  v += __shfl_xor(v, 1, 16);
  v += __shfl_xor(v, 2, 16);
  v += __shfl_xor(v, 4, 16);
  v += __shfl_xor(v, 8, 16);
  return v;
}

__global__ __launch_bounds__(NT) void twostream_scan(
    const float* __restrict__ tD,  const float* __restrict__ tF,
    const float* __restrict__ esD, const float* __restrict__ esF,
    const float* __restrict__ rsD, const float* __restrict__ rsF,
    const float* __restrict__ fdD0,const float* __restrict__ fdF0,
    float* __restrict__ out)
{
  const int c = threadIdx.x;                 // 0..47 channel
  const int y = threadIdx.y;                 // 0..NB-1
  const int t = c + Cn * y;                  // 0..95 linear tid
  const int b = blockIdx.x * NB + y;         // batch index

  // Per-layer carry checkpoints (rb_dir, rb_dif): [layer][tid] -> bank-conflict-free b64.
  __shared__ float2 rbbuf[Ln][NT];           // 45 KB
  __shared__ float  part[6][4];              // cross-wave partial sums
#if ASYNC_LDS
  __shared__ float  stage[2][8][NT];         // double-buffered input staging, 6 KB
  int cur = 0;
  auto issue = [&](int buf, int l) {
    const int g0 = (b * Ln + l) * Cn + c;
    const int eb = g0 * 3;
    __builtin_amdgcn_global_load_async_to_lds_b32((Gint*)(tD  + g0    ), (Lint*)&stage[buf][0][t], 0, 0);
    __builtin_amdgcn_global_load_async_to_lds_b32((Gint*)(tF  + g0    ), (Lint*)&stage[buf][1][t], 0, 0);
    __builtin_amdgcn_global_load_async_to_lds_b32((Gint*)(esD + eb + 0), (Lint*)&stage[buf][2][t], 0, 0);
    __builtin_amdgcn_global_load_async_to_lds_b32((Gint*)(esD + eb + 1), (Lint*)&stage[buf][3][t], 0, 0);
    __builtin_amdgcn_global_load_async_to_lds_b32((Gint*)(esD + eb + 2), (Lint*)&stage[buf][4][t], 0, 0);
    __builtin_amdgcn_global_load_async_to_lds_b32((Gint*)(esF + eb + 0), (Lint*)&stage[buf][5][t], 0, 0);
    __builtin_amdgcn_global_load_async_to_lds_b32((Gint*)(esF + eb + 1), (Lint*)&stage[buf][6][t], 0, 0);
    __builtin_amdgcn_global_load_async_to_lds_b32((Gint*)(esF + eb + 2), (Lint*)&stage[buf][7][t], 0, 0);
  };
#endif

  float td, tf, etD, erD, eaD, etF, erF, eaF;
  auto fetch = [&](int l, int nextl) {
#if ASYNC_LDS
    WAIT_ASYNC0();
    td  = stage[cur][0][t]; tf  = stage[cur][1][t];
    etD = stage[cur][2][t]; erD = stage[cur][3][t]; eaD = stage[cur][4][t];
    etF = stage[cur][5][t]; erF = stage[cur][6][t]; eaF = stage[cur][7][t];
    if (nextl >= 0) issue(cur ^ 1, nextl);     // one-layer lookahead
    cur ^= 1;
#else
    const int g0 = (b * Ln + l) * Cn + c;
    const int eb = g0 * 3;
    td  = tD[g0];      tf  = tF[g0];
    etD = esD[eb];     erD = esD[eb + 1]; eaD = esD[eb + 2];
    etF = esF[eb];     erF = esF[eb + 1]; eaF = esF[eb + 2];
    if (nextl >= 0) {
      const int n0 = (b * Ln + nextl) * Cn + c;
      __builtin_prefetch(tD  + n0,     0, 0);  // global_prefetch_b8
      __builtin_prefetch(tF  + n0,     0, 0);
      __builtin_prefetch(esD + n0 * 3, 0, 0);
      __builtin_prefetch(esF + n0 * 3, 0, 0);
    }
#endif
  };

  // Reduce 4 values over the 48 channels of each batch-in-block, store to out.
  auto reduce_store = [&](float v0, float v1, float v2, float v3, int j, bool withAbs) {
    const float s0 = red16(v0), s1 = red16(v1), s2 = red16(v2), s3 = red16(v3);
    const int g = t >> 4;                      // 16-lane group id; groups 3y..3y+2 cover batch y
    if ((t & 15) == 0) { part[g][0] = s0; part[g][1] = s1; part[g][2] = s2; part[g][3] = s3; }
    __syncthreads();
    if (t < 4 * NB) {
      const int yy = t >> 2, k = t & 3;
      const int bb = blockIdx.x * NB + yy;
      const float s = part[3 * yy][k] + part[3 * yy + 1][k] + part[3 * yy + 2][k];
      if (k < 3)        out[k * (Bn * (Ln + 1)) + bb * (Ln + 1) + j] = s;
      else if (withAbs) out[3 * (Bn * (Ln + 1)) + bb * Ln + (j - 1)] = s;
    }
    __syncthreads();
  };

  // -------- upward scan (surface -> top), checkpoint carry per layer --------
  float rb_d = rsD[b * Cn + c];
  float rb_f = rsF[b * Cn + c];
  float ab_d = 1.f - rb_d, ab_f = 1.f - rb_f;

#if ASYNC_LDS
  issue(cur, Ln - 1);
#endif
  for (int l = Ln - 1; l >= 0; --l) {
    rbbuf[l][t] = make_float2(rb_d, rb_f);     // carry BELOW layer l
    fetch(l, l - 1);
    const float e_d = 1.f - td, e_f = 1.f - tf;
    const float d    = 1.f / (1.f - e_f * erF * rb_f + EPSF);
    const float tm_d = td * rb_d * e_f * erF * d + e_d * etD * d;
    const float abm_d = td * ab_d + tm_d * ab_f;
    const float rbm_d = td * rb_d * d + e_d * etD * rb_f * d;
    const float atm_d = e_d * eaD + rbm_d * e_f * eaF;
    const float rm_d  = e_d * erD + rbm_d * (tf + e_f * etF);
    const float tm_f  = tf * rb_f * e_f * erF * d + e_f * etF * d;
    const float abm_f = tf * ab_f + tm_f * ab_f;
    const float rbm_f = tf * rb_f * d + e_f * etF * rb_f * d;
    const float atm_f = e_f * eaF + rbm_f * e_f * eaF;
    const float rm_f  = e_f * erF + rbm_f * (tf + e_f * etF);
    rb_d = rm_d;            rb_f = rm_f;
    ab_d = atm_d + abm_d;   ab_f = atm_f + abm_f;
  }
  const float rmulti = rb_d;                   // r_multi_direct

  // -------- downward scan (top -> surface), recompute layer quantities --------
  float fd_d = fdD0[b * Cn + c];
  float fd_f = fdF0[b * Cn + c];

#if ASYNC_LDS
  issue(cur, 0);
#endif
  reduce_store(fd_d, fd_f, fd_d * rmulti, 0.f, 0, false);   // level-0 outputs

  for (int l = 0; l < Ln; ++l) {
    const float2 rb = rbbuf[l][t];
    fetch(l, (l + 1 < Ln) ? l + 1 : -1);
    const float e_d = 1.f - td, e_f = 1.f - tf;
    const float d    = 1.f / (1.f - e_f * erF * rb.y + EPSF);
    const float tm_d  = td * rb.x * e_f * erF * d + e_d * etD * d;
    const float rbm_d = td * rb.x * d + e_d * etD * rb.y * d;
    const float atm_d = e_d * eaD + rbm_d * e_f * eaF;
    const float tm_f  = tf * rb.y * e_f * erF * d + e_f * etF * d;
    const float rbm_f = tf * rb.y * d + e_f * etF * rb.y * d;
    const float atm_f = e_f * eaF + rbm_f * e_f * eaF;

    const float absorbed = fd_d * atm_d + fd_f * atm_f;
    const float nd  = fd_d * td;
    const float ndf = fd_d * tm_d + fd_f * (tf + tm_f);
    const float fu  = fd_d * rbm_d + fd_f * rbm_f;
    fd_d = nd; fd_f = ndf;

    reduce_store(nd, ndf, fu, absorbed, l + 1, true);
  }
}

extern "C" void kernel_launch(void* const* d_in, const int* in_sizes, int n_in,
                              void* d_out, int out_size, void* d_ws, size_t ws_size,
                              hipStream_t stream) {
  (void)in_sizes; (void)n_in; (void)d_ws; (void)ws_size; (void)out_size;
  const float* tD   = (const float*)d_in[0];
  const float* tF   = (const float*)d_in[1];
  const float* esD  = (const float*)d_in[2];
  const float* esF  = (const float*)d_in[3];
  const float* rsD  = (const float*)d_in[4];
  const float* rsF  = (const float*)d_in[5];
  const float* fdD0 = (const float*)d_in[6];
  const float* fdF0 = (const float*)d_in[7];
  float* out = (float*)d_out;

  dim3 block(Cn, NB, 1);      // 96 threads = 3 full wave32s
  dim3 grid(Bn / NB, 1, 1);   // 1024 blocks
  twostream_scan<<<grid, block, 0, stream>>>(tD, tF, esD, esF, rsD, rsF, fdD0, fdF0, out);
}